// DAS_FT_46170898432508
// MI455X (gfx1250) — compile-verified
//
#include <hip/hip_runtime.h>
#include <hip/hip_bf16.h>

typedef __attribute__((ext_vector_type(2))) float v2f;
typedef __attribute__((ext_vector_type(8))) float v8f;

#define DAS_MIN_WIDTH 0.001f

// Raw v_sqrt_f32 (~1 ulp), skips the IEEE correctly-rounded fixup sequence.
__device__ __forceinline__ float fast_sqrtf(float x) {
    return __builtin_amdgcn_sqrtf(x);
}

struct DasCtx {
    float px, py, pz, txdel;
    float fs_c, tz0fs, phi_s, phi_b;
    float rxfnum, ex0, exN;
    int S;
    unsigned rowbase;            // t*E
    const float* idata;
    const float* qdata;
    const float4* lds_ele4;      // ele_pos padded to 16B stride in LDS
};

// One (z,e) term: geometry -> delay -> interp gather -> demod rotate -> apod.
template <bool TAIL>
__device__ __forceinline__ void das_term(const DasCtx& cx, int e, int E,
                                         float& out_i, float& out_q) {
    const bool evalid = !TAIL || (e < E);
    const int  ec = TAIL ? (evalid ? e : (E - 1)) : e;

    // Single aligned ds_load_b128 broadcast (all 16 lanes of a half-wave
    // read the same element).
    const float4 ep = cx.lds_ele4[ec];
    const float vx = cx.px - ep.x;
    const float vy = cx.py - ep.y;
    const float vz = cx.pz - ep.z;
    const float rxdel = fast_sqrtf(vx * vx + vy * vy + vz * vz);

    const float s      = cx.txdel + rxdel;
    const float delays = s * cx.fs_c + cx.tz0fs;
    const float i0f  = floorf(delays);
    const float frac = delays - i0f;
    const int   i0   = (int)i0f;
    const int   i1   = i0 + 1;
    const bool  ok0  = (i0 >= 0) && (i0 < cx.S);
    const bool  ok1  = (i1 >= 0) && (i1 < cx.S);
    const int   c0   = min(max(i0, 0), cx.S - 1);
    const int   c1   = min(max(i1, 0), cx.S - 1);

    // Branchless clamped gathers (keeps EXEC full for the WMMAs).
    // Element index fits u32 -> base+voffset scale_offset addressing.
    const unsigned row = (cx.rowbase + (unsigned)ec) * (unsigned)cx.S;
    const float id0 = cx.idata[row + (unsigned)c0] * (ok0 ? 1.0f : 0.0f);
    const float id1 = cx.idata[row + (unsigned)c1] * (ok1 ? 1.0f : 0.0f);
    const float qd0 = cx.qdata[row + (unsigned)c0] * (ok0 ? 1.0f : 0.0f);
    const float qd1 = cx.qdata[row + (unsigned)c1] * (ok1 ? 1.0f : 0.0f);

    const float omf  = 1.0f - frac;
    const float ifoc = omf * id0 + frac * id1;
    const float qfoc = omf * qd0 + frac * qd1;

    // Demodulation: theta = pi*phi2; sincospif does an exact mod-2 reduction.
    // fdemod==0 -> phi2==0 -> identity rotation.
    const float phi2 = s * cx.phi_s + cx.phi_b;
    float st, ct;
    sincospif(phi2, &st, &ct);
    const float ri = ifoc * ct - qfoc * st;
    const float rq = qfoc * ct + ifoc * st;

    // Apodization mask (mask TRUE -> apod = 1). IEEE division kept so the
    // discrete accept/reject decision is bit-identical to the reference.
    bool mask = (fabsf(vz / vx) > cx.rxfnum);
    mask = mask || (fabsf(vx) <= DAS_MIN_WIDTH);
    mask = mask || ((vx >=  DAS_MIN_WIDTH) && (cx.px <= cx.ex0));
    mask = mask || ((vx <= -DAS_MIN_WIDTH) && (cx.px >= cx.exN));
    const float apod = (mask && evalid) ? 1.0f : 0.0f;

    out_i = ri * apod;
    out_q = rq * apod;
}

// One wave computes 16 consecutive z-pixels for one transmit t, reducing over
// all E receive elements with chained V_WMMA_F32_16X16X4_F32 (B = ones):
//   D[m][n] = sum_k A[m][k] + C[m][n]
// A-layout (16x4 f32, ISA 7.12.2): lanes 0-15 -> M=lane, VGPR0=K0, VGPR1=K1;
// lanes 16-31 -> M=lane-16, VGPR0=K2, VGPR1=K3. Each K=4 step covers 4
// elements e; each lane evaluates 2 (z,e) terms. Out-of-range z rows are
// isolated (only their own D row) and never stored, so no z masking needed.
__global__ __launch_bounds__(256) void das_wmma_kernel(
    const float* __restrict__ idata, const float* __restrict__ qdata,
    const float* __restrict__ grid,  const float* __restrict__ tx_ori,
    const float* __restrict__ ele_pos, const float* __restrict__ time_zero,
    const float* __restrict__ fsp, const float* __restrict__ cp,
    const float* __restrict__ fdp, const float* __restrict__ fnp,
    float* __restrict__ out,
    int T, int E, int S, int Z)
{
    extern __shared__ float lds_ele[];   // 4*E floats (16B-stride padded)

    // Stage element positions into LDS once, padded to float4 stride.
    for (int i = threadIdx.x; i < E; i += blockDim.x) {
        lds_ele[4 * i + 0] = ele_pos[3 * i + 0];
        lds_ele[4 * i + 1] = ele_pos[3 * i + 1];
        lds_ele[4 * i + 2] = ele_pos[3 * i + 2];
        lds_ele[4 * i + 3] = 0.0f;
    }
    __syncthreads();

    const int lane  = threadIdx.x & 31;
    const int wave  = threadIdx.x >> 5;
    const int t     = blockIdx.y;
    const int zbase = blockIdx.x * 128 + wave * 16;
    const int m     = lane & 15;

    int z = zbase + m;
    if (z >= Z) z = Z - 1;   // clamped; this row is never stored

    DasCtx cx;
    cx.S = S;
    cx.rowbase = (unsigned)t * (unsigned)E;
    cx.idata = idata;
    cx.qdata = qdata;
    cx.lds_ele4 = (const float4*)lds_ele;

    // Scalars (1-element device arrays)
    const float fs     = fsp[0];
    const float c      = cp[0];
    const float fdemod = fdp[0];
    cx.rxfnum = fnp[0];
    const float tz0 = time_zero[t];
    cx.ex0 = ele_pos[0];
    cx.exN = ele_pos[(size_t)(E - 1) * 3];

    // Per-(t,z) geometry, fixed across the whole e-loop.
    const size_t gidx = ((size_t)t * Z + z) * 3;
    cx.px = grid[gidx + 0];
    cx.py = grid[gidx + 1];
    cx.pz = grid[gidx + 2];
    const float ox = tx_ori[(size_t)t * 3 + 0];
    const float oy = tx_ori[(size_t)t * 3 + 1];
    const float oz = tx_ori[(size_t)t * 3 + 2];
    const float tdx = cx.px - ox, tdy = cx.py - oy, tdz = cx.pz - oz;
    cx.txdel = fast_sqrtf(tdx * tdx + tdy * tdy + tdz * tdz);

    // Folded constants: delays = s*fs_c + tz0fs ; phi2 = s*phi_s + phi_b,
    // where s = txdel + rxdel and sin/cos(pi*phi2) == sin/cos(theta).
    const float invc = 1.0f / c;
    cx.fs_c  = fs * invc;
    cx.tz0fs = tz0 * fs;
    const float gz2c = cx.pz * 2.0f * invc;
    cx.phi_s = 2.0f * fdemod * invc;
    cx.phi_b = 2.0f * fdemod * (tz0 - gz2c);

    const int eoff = (lane >> 4) << 1;  // 0 for lanes 0-15, 2 for lanes 16-31

    v8f ci = {0.f, 0.f, 0.f, 0.f, 0.f, 0.f, 0.f, 0.f};
    v8f cq = {0.f, 0.f, 0.f, 0.f, 0.f, 0.f, 0.f, 0.f};
    const v2f ones = {1.0f, 1.0f};

    // Main loop: E multiple-of-4 part, no element clamping in the hot path.
    const int nfull = E >> 2;
#pragma unroll 2
    for (int step = 0; step < nfull; ++step) {
        float a0i, a0q, a1i, a1q;
        das_term<false>(cx, step * 4 + eoff + 0, E, a0i, a0q);
        das_term<false>(cx, step * 4 + eoff + 1, E, a1i, a1q);
        v2f ai, aq;
        ai[0] = a0i; ai[1] = a1i;
        aq[0] = a0q; aq[1] = a1q;
        // 8 args: (neg_a, A, neg_b, B, c_mod, C, reuse_a, reuse_b)
        ci = __builtin_amdgcn_wmma_f32_16x16x4_f32(false, ai, false, ones,
                                                   (short)0, ci, false, false);
        cq = __builtin_amdgcn_wmma_f32_16x16x4_f32(false, aq, false, ones,
                                                   (short)0, cq, false, false);
    }
    // Tail step (E % 4 != 0): clamped, invalid e contribute zero.
    if (E & 3) {
        float a0i, a0q, a1i, a1q;
        das_term<true>(cx, nfull * 4 + eoff + 0, E, a0i, a0q);
        das_term<true>(cx, nfull * 4 + eoff + 1, E, a1i, a1q);
        v2f ai, aq;
        ai[0] = a0i; ai[1] = a1i;
        aq[0] = a0q; aq[1] = a1q;
        ci = __builtin_amdgcn_wmma_f32_16x16x4_f32(false, ai, false, ones,
                                                   (short)0, ci, false, false);
        cq = __builtin_amdgcn_wmma_f32_16x16x4_f32(false, aq, false, ones,
                                                   (short)0, cq, false, false);
    }

    // Extract: lanes 0-15 hold D[r][lane] in c[r]; lanes 16-31 hold
    // D[8+r][lane-16]. Diagonal pick -> 16 writer lanes cover 16 z rows.
    const int sel = lane & 7;
    float vi = ci[0], vq = cq[0];
#pragma unroll
    for (int j = 1; j < 8; ++j) {
        if (sel == j) { vi = ci[j]; vq = cq[j]; }
    }
    const bool lowW = (lane < 8);                    // rows 0..7
    const bool hiW  = (lane >= 16) && (lane < 24);   // rows 8..15
    const int  zo   = zbase + (lowW ? sel : (8 + sel));
    if ((lowW || hiW) && zo < Z) {
        const size_t TZ = (size_t)T * (size_t)Z;
        out[(size_t)t * Z + zo]      = vi;  // idas
        out[TZ + (size_t)t * Z + zo] = vq;  // qdas
    }
}

extern "C" void kernel_launch(void* const* d_in, const int* in_sizes, int n_in,
                              void* d_out, int out_size, void* d_ws, size_t ws_size,
                              hipStream_t stream) {
    const float* idata    = (const float*)d_in[0];
    const float* qdata    = (const float*)d_in[1];
    const float* grid     = (const float*)d_in[2];
    const float* tx_ori   = (const float*)d_in[3];
    const float* ele_pos  = (const float*)d_in[4];
    const float* time_z   = (const float*)d_in[5];
    const float* fsp      = (const float*)d_in[6];
    const float* cp       = (const float*)d_in[7];
    const float* fdp      = (const float*)d_in[8];
    const float* fnp      = (const float*)d_in[9];

    const int T = in_sizes[3] / 3;                 // tx_ori: (T,3)
    const int E = in_sizes[4] / 3;                 // ele_pos: (E,3)
    const int Z = in_sizes[2] / (T * 3);           // grid: (T,Z,3)
    const int S = in_sizes[0] / (T * E);           // idata: (T,E,S)

    dim3 block(256);                               // 8 waves x 16 z = 128 z/block
    dim3 gridDim((Z + 127) / 128, T);
    const size_t shmem = (size_t)(4 * E) * sizeof(float);
    das_wmma_kernel<<<gridDim, block, shmem, stream>>>(
        idata, qdata, grid, tx_ori, ele_pos, time_z,
        fsp, cp, fdp, fnp, (float*)d_out, T, E, S, Z);
}